// DynamicIsotropicEKF0_12180527251750
// MI455X (gfx1250) — compile-verified
//
#include <hip/hip_runtime.h>
#include <math.h>

// CDNA5 / gfx1250. wave32. FP32 WMMA (V_WMMA_F32_16X16X4_F32) used for the
// 5x5 preconditioned transition matrix applied to the (5, D) state.

typedef __attribute__((ext_vector_type(2))) float v2f;
typedef __attribute__((ext_vector_type(8))) float v8f;

#define NDIM 5

__device__ __forceinline__ void compute_precond(float adt, float* p, float* pinv) {
    // powers (descending): 4,3,2,1,0 ; scales = k! ; powers += 0.5
    const float scales[NDIM] = {24.f, 6.f, 2.f, 1.f, 1.f};
    #pragma unroll
    for (int i = 0; i < NDIM; ++i) {
        float pw = (float)(4 - i) + 0.5f;
        float ap = powf(adt, pw);
        p[i]    = ap / scales[i];
        pinv[i] = scales[i] / ap;
    }
}

// ---------------------------------------------------------------------------
// Pass 1: m_ext = diag(p) * A * diag(pinv) * m0   via f32 WMMA (16x16x4),
// plus per-block partial sums of bias^2 where bias = m_ext[1] - sin(m_ext[0]).
// ---------------------------------------------------------------------------
__global__ void k_extrapolate_mean(const float* __restrict__ m0,
                                   const float* __restrict__ a,
                                   const float* __restrict__ dtp,
                                   float* __restrict__ m_ext,
                                   float* __restrict__ partials,
                                   int D) {
    __shared__ float red[256];

    const int lane  = threadIdx.x & 31;
    const int row16 = lane & 15;   // M index (rows 0..15 of the 16x16 tile)
    const int hi    = lane >> 4;   // which K-pair this lane holds

    float p[NDIM], pinv[NDIM];
    compute_precond(fabsf(dtp[0]), p, pinv);

    // A fragment: M = diag(p) * a * diag(pinv), padded to 16 rows.
    // 32-bit A 16x4 layout: lanes 0-15 hold {K=0,K=1}, lanes 16-31 hold {K=2,K=3}.
    v2f afrag0 = {0.f, 0.f};   // K columns 0..3
    v2f afrag1 = {0.f, 0.f};   // K column 4 (zero-padded to K=0..3; only K=0 used)
    if (row16 < NDIM) {
        const int k0 = 2 * hi, k1 = 2 * hi + 1;
        const float pr = p[row16];
        afrag0.x = pr * a[row16 * NDIM + k0] * pinv[k0];
        afrag0.y = pr * a[row16 * NDIM + k1] * pinv[k1];
        if (hi == 0) afrag1.x = pr * a[row16 * NDIM + 4] * pinv[4];
    }

    const long long Dll = (long long)D;
    const int numTiles = (D + 15) >> 4;
    const int waveId   = (int)((blockIdx.x * blockDim.x + threadIdx.x) >> 5);
    const int numWaves = (int)((gridDim.x * blockDim.x) >> 5);

    float lsum = 0.f;
    for (int t = waveId; t < numTiles; t += numWaves) {   // wave-uniform bound
        const int  col = (t << 4) + row16;
        const bool inb = (col < D);

        // B fragment (4x16): lanes 0-15 hold rows K=0,1; lanes 16-31 rows K=2,3.
        v2f bfrag0 = {0.f, 0.f};
        v2f bfrag1 = {0.f, 0.f};
        if (inb) {
            bfrag0.x = m0[(long long)(2 * hi)     * Dll + col];
            bfrag0.y = m0[(long long)(2 * hi + 1) * Dll + col];
            if (hi == 0) bfrag1.x = m0[4ll * Dll + col];
        }

        v8f acc = {};
        acc = __builtin_amdgcn_wmma_f32_16x16x4_f32(false, afrag0, false, bfrag0,
                                                    (short)0, acc, false, false);
        acc = __builtin_amdgcn_wmma_f32_16x16x4_f32(false, afrag1, false, bfrag1,
                                                    (short)0, acc, false, false);

        // C/D layout: lanes 0-15 hold N=lane, VGPR v holds row M=v (v=0..7).
        if (hi == 0 && inb) {
            const float e0 = acc[0];
            const float e1 = acc[1];
            m_ext[0ll * Dll + col] = e0;
            m_ext[1ll * Dll + col] = e1;
            m_ext[2ll * Dll + col] = acc[2];
            m_ext[3ll * Dll + col] = acc[3];
            m_ext[4ll * Dll + col] = acc[4];
            const float bias = e1 - sinf(e0);
            lsum += bias * bias;
        }
    }

    // deterministic fixed-topology block reduction
    red[threadIdx.x] = lsum;
    __syncthreads();
    for (int s = 128; s > 0; s >>= 1) {
        if ((int)threadIdx.x < s) red[threadIdx.x] += red[threadIdx.x + s];
        __syncthreads();
    }
    if (threadIdx.x == 0) partials[blockIdx.x] = red[0];
}

// ---------------------------------------------------------------------------
// Pass 2 (single block): finish reduction, QR combine, gain, small outputs.
// ---------------------------------------------------------------------------
__global__ void k_small(const float* __restrict__ a,
                        const float* __restrict__ c_sqrtm0,
                        const float* __restrict__ qup,
                        const float* __restrict__ dtp,
                        const float* __restrict__ partials, int nPartials,
                        long long D,
                        float* __restrict__ out_ccor,
                        float* __restrict__ out_cext,
                        float* __restrict__ out_err,
                        float* __restrict__ g_ws) {
    __shared__ float red[256];
    float s = 0.f;
    for (int i = threadIdx.x; i < nPartials; i += blockDim.x) s += partials[i];
    red[threadIdx.x] = s;
    __syncthreads();
    for (int st = 128; st > 0; st >>= 1) {
        if ((int)threadIdx.x < st) red[threadIdx.x] += red[threadIdx.x + st];
        __syncthreads();
    }
    if (threadIdx.x != 0) return;

    const float sumsq = red[0];
    const float dt = dtp[0];
    float p[NDIM], pinv[NDIM];
    compute_precond(fabsf(dt), p, pinv);

    // s_scal = || pinv[1] * q_lower[1,:] ||^2 ; q_lower = qup^T
    float s_scal = 0.f;
    #pragma unroll
    for (int k = 0; k < NDIM; ++k) {
        const float v = pinv[1] * qup[k * NDIM + 1];
        s_scal += v * v;
    }
    const float diffusion = sqrtf(sumsq / ((float)D * s_scal));
    const float err = dt * diffusion * sqrtf(s_scal);

    // T = a @ (pinv[:,None] * c_sqrtm0)
    float T[NDIM][NDIM];
    for (int i = 0; i < NDIM; ++i)
        for (int j = 0; j < NDIM; ++j) {
            float acc = 0.f;
            for (int k = 0; k < NDIM; ++k)
                acc += a[i * NDIM + k] * pinv[k] * c_sqrtm0[k * NDIM + j];
            T[i][j] = acc;
        }

    // X = [ R1 ; R2 ]  (10x5) ; R1 = T^T ; R2 = diffusion * qup^T
    float X[2 * NDIM][NDIM];
    for (int i = 0; i < NDIM; ++i)
        for (int j = 0; j < NDIM; ++j) {
            X[i][j]        = T[j][i];
            X[NDIM + i][j] = diffusion * qup[j * NDIM + i];
        }

    // Householder QR (LAPACK sign convention): R = upper triangle of X
    for (int k = 0; k < NDIM; ++k) {
        float nrm2 = 0.f;
        for (int i = k; i < 2 * NDIM; ++i) nrm2 += X[i][k] * X[i][k];
        const float nrm = sqrtf(nrm2);
        if (nrm > 0.f) {
            const float alpha = (X[k][k] >= 0.f) ? -nrm : nrm;
            float v[2 * NDIM];
            for (int i = k; i < 2 * NDIM; ++i) v[i] = X[i][k];
            v[k] -= alpha;
            float vn2 = 0.f;
            for (int i = k; i < 2 * NDIM; ++i) vn2 += v[i] * v[i];
            if (vn2 > 0.f) {
                for (int j = k; j < NDIM; ++j) {
                    float dot = 0.f;
                    for (int i = k; i < 2 * NDIM; ++i) dot += v[i] * X[i][j];
                    const float f = 2.f * dot / vn2;
                    for (int i = k; i < 2 * NDIM; ++i) X[i][j] -= f * v[i];
                }
            }
        }
    }

    // c_ext = diag(p) * R^T
    float cext[NDIM][NDIM];
    for (int i = 0; i < NDIM; ++i)
        for (int j = 0; j < NDIM; ++j) {
            const float r = (j <= i) ? X[j][i] : 0.f;   // R[j][i]
            cext[i][j] = p[i] * r;
            out_cext[i * NDIM + j] = cext[i][j];
        }

    // correction gain
    float ssq[NDIM];
    float S = 0.f;
    #pragma unroll
    for (int k = 0; k < NDIM; ++k) { ssq[k] = cext[k][1]; S += ssq[k] * ssq[k]; }
    float g[NDIM];
    for (int i = 0; i < NDIM; ++i) {
        float acc = 0.f;
        for (int k = 0; k < NDIM; ++k) acc += cext[k][i] * ssq[k];
        g[i] = acc / S;
        g_ws[i] = g[i];
    }

    // c_cor[i][j] = cext[i][j] - g[j] * ssq[i]
    for (int i = 0; i < NDIM; ++i)
        for (int j = 0; j < NDIM; ++j)
            out_ccor[i * NDIM + j] = cext[i][j] - g[j] * ssq[i];

    out_err[0] = err;
}

// ---------------------------------------------------------------------------
// Pass 3: m_cor = m_ext - g[:,None]*bias ; u = m_cor[0]. Fully coalesced.
// ---------------------------------------------------------------------------
__global__ void k_correct(const float* __restrict__ m_ext,
                          const float* __restrict__ g_ws,
                          float* __restrict__ m_cor,
                          float* __restrict__ u,
                          long long D) {
    const long long i = (long long)blockIdx.x * blockDim.x + threadIdx.x;
    if (i >= D) return;
    const float e0 = m_ext[0 * D + i];
    const float e1 = m_ext[1 * D + i];
    const float e2 = m_ext[2 * D + i];
    const float e3 = m_ext[3 * D + i];
    const float e4 = m_ext[4 * D + i];
    const float bias = e1 - sinf(e0);
    const float c0 = e0 - g_ws[0] * bias;
    m_cor[0 * D + i] = c0;
    m_cor[1 * D + i] = e1 - g_ws[1] * bias;
    m_cor[2 * D + i] = e2 - g_ws[2] * bias;
    m_cor[3 * D + i] = e3 - g_ws[3] * bias;
    m_cor[4 * D + i] = e4 - g_ws[4] * bias;
    u[i] = c0;
}

extern "C" void kernel_launch(void* const* d_in, const int* in_sizes, int n_in,
                              void* d_out, int out_size, void* d_ws, size_t ws_size,
                              hipStream_t stream) {
    const float* m0       = (const float*)d_in[0];
    const float* c_sqrtm0 = (const float*)d_in[1];
    const float* a        = (const float*)d_in[2];
    const float* qup      = (const float*)d_in[3];
    const float* dt       = (const float*)d_in[4];

    const long long D = (long long)(in_sizes[0] / NDIM);

    // output layout: m_cor(5D) | c_cor(25) | m_ext(5D) | c_ext(25) | err(1) | u(D)
    float* out    = (float*)d_out;
    float* m_cor  = out;
    float* c_cor  = out + 5 * D;
    float* m_ext  = c_cor + NDIM * NDIM;
    float* c_ext  = m_ext + 5 * D;
    float* errp   = c_ext + NDIM * NDIM;
    float* u      = errp + 1;

    float* ws       = (float*)d_ws;
    float* g_ws     = ws;       // 5 floats
    float* partials = ws + 8;   // G1 floats

    const int G1 = 1024, T1 = 256;
    k_extrapolate_mean<<<G1, T1, 0, stream>>>(m0, a, dt, m_ext, partials, (int)D);

    k_small<<<1, 256, 0, stream>>>(a, c_sqrtm0, qup, dt, partials, G1, D,
                                   c_cor, c_ext, errp, g_ws);

    const int T3 = 256;
    const int G3 = (int)((D + T3 - 1) / T3);
    k_correct<<<G3, T3, 0, stream>>>(m_ext, g_ws, m_cor, u, D);
}